// PerturbNet_60911226192062
// MI455X (gfx1250) — compile-verified
//
#include <hip/hip_runtime.h>
#include <math.h>

#define BLOCK 256
#define MPB   256   // models per block (1 per thread), 32 per wave

// Per-wave LDS region: 32 models, floats:
//   W2 : 32*25 = 800 @ 0     (bytes    0..3199)
//   W1 : 160         @ 800   (bytes 3200..3839)
//   b1 : 160         @ 960   (bytes 3840..4479)
//   b2 : 160         @ 1120  (bytes 4480..5119)
//   W3 : 160         @ 1280  (bytes 5120..5759)
// 1440 floats = 5760 bytes per wave; 8 waves -> 46080 bytes
#define WAVE_FLOATS 1440
#define SMEM_FLOATS (WAVE_FLOATS * 8)

// gfx1250 async global->LDS copy; IOFFSET applies to BOTH LDS dest and global src
#define ACP(ldsaddr, gaddr, OFF)                                            \
    asm volatile("global_load_async_to_lds_b128 %0, %1, off offset:" OFF    \
                 :: "v"(ldsaddr), "v"(gaddr) : "memory")

__device__ __forceinline__ float elu_fast(float v) {
    // matches jax.nn.elu to ~2^-24 absolute; v_exp_f32 hardware path
    return v > 0.0f ? v : __expf(v) - 1.0f;
}

__global__ void __launch_bounds__(BLOCK)
perturbnet_kernel(const float* __restrict__ x,
                  const float* __restrict__ W1,
                  const float* __restrict__ b1,
                  const float* __restrict__ W2,
                  const float* __restrict__ b2,
                  const float* __restrict__ W3,
                  const float* __restrict__ b3,
                  float* __restrict__ out,
                  int n) {
    __shared__ __align__(16) float smem[SMEM_FLOATS];

    const int tid  = (int)threadIdx.x;
    const int lane = tid & 31;
    const int w    = tid >> 5;
    const int m0   = (int)blockIdx.x * MPB;
    const int mw   = m0 + w * 32;        // first model of this wave
    const int m    = mw + lane;
    const bool full = (m0 + MPB) <= n;   // block-uniform

    if (full) {
        const unsigned lds_wave = (unsigned)(unsigned long long)(uintptr_t)smem
                                + (unsigned)w * 5760u;
        const unsigned lb = (unsigned)lane * 16u;

        const char* gW2 = (const char*)(W2 + (size_t)mw * 25) + (size_t)lane * 16;
        const char* gW1 = (const char*)(W1 + (size_t)mw * 5)  + (size_t)lane * 16;
        const char* gb1 = (const char*)(b1 + (size_t)mw * 5)  + (size_t)lane * 16;
        const char* gb2 = (const char*)(b2 + (size_t)mw * 5)  + (size_t)lane * 16;
        const char* gW3 = (const char*)(W3 + (size_t)mw * 5)  + (size_t)lane * 16;

        const unsigned ldsW2 = lds_wave +    0u + lb;
        const unsigned ldsW1 = lds_wave + 3200u + lb;
        const unsigned ldsb1 = lds_wave + 3840u + lb;
        const unsigned ldsb2 = lds_wave + 4480u + lb;
        const unsigned ldsW3 = lds_wave + 5120u + lb;

        // W2: 200 float4 per wave = 6 full rounds of 32 lanes + 8-lane tail
        ACP(ldsW2, gW2, "0");
        ACP(ldsW2, gW2, "512");
        ACP(ldsW2, gW2, "1024");
        ACP(ldsW2, gW2, "1536");
        ACP(ldsW2, gW2, "2048");
        ACP(ldsW2, gW2, "2560");
        // small arrays: 40 float4 each = 1 full round + 8-lane tail
        ACP(ldsW1, gW1, "0");
        ACP(ldsb1, gb1, "0");
        ACP(ldsb2, gb2, "0");
        ACP(ldsW3, gW3, "0");
        if (lane < 8) {
            ACP(ldsW2, gW2, "3072");
            ACP(ldsW1, gW1, "512");
            ACP(ldsb1, gb1, "512");
            ACP(ldsb2, gb2, "512");
            ACP(ldsW3, gW3, "512");
        }

        // overlap: coalesced per-lane loads while async copies are in flight
        const float xv  = x[m];
        const float b3v = b3[m];

        // wave-private staging: only this wave reads what it wrote -> no __syncthreads
        asm volatile("s_wait_asynccnt 0" ::: "memory");

        const float* base = smem + w * WAVE_FLOATS;
        const float* lw2  = base +        lane * 25;
        const float* lw1  = base +  800 + lane * 5;
        const float* lb1p = base +  960 + lane * 5;
        const float* lb2p = base + 1120 + lane * 5;
        const float* lw3  = base + 1280 + lane * 5;

        float h1[5];
#pragma unroll
        for (int j = 0; j < 5; ++j)
            h1[j] = elu_fast(fmaf(xv, lw1[j], lb1p[j]));

        float h2[5];
#pragma unroll
        for (int o = 0; o < 5; ++o) {
            float acc = lb2p[o];
#pragma unroll
            for (int h = 0; h < 5; ++h)
                acc = fmaf(h1[h], lw2[o * 5 + h], acc);
            h2[o] = elu_fast(acc);
        }

        float acc = b3v;
#pragma unroll
        for (int o = 0; o < 5; ++o)
            acc = fmaf(h2[o], lw3[o], acc);

        out[m] = acc;
    } else {
        // tail block (not hit for N = 2^20): direct global path, guarded
        if (m < n) {
            const float xv = x[m];
            const float* gw1 = W1 + (size_t)m * 5;
            const float* gb1 = b1 + (size_t)m * 5;
            float h1[5];
#pragma unroll
            for (int j = 0; j < 5; ++j)
                h1[j] = elu_fast(fmaf(xv, gw1[j], gb1[j]));

            const float* gw2 = W2 + (size_t)m * 25;
            const float* gb2 = b2 + (size_t)m * 5;
            float h2[5];
#pragma unroll
            for (int o = 0; o < 5; ++o) {
                float acc = gb2[o];
#pragma unroll
                for (int h = 0; h < 5; ++h)
                    acc = fmaf(h1[h], gw2[o * 5 + h], acc);
                h2[o] = elu_fast(acc);
            }

            const float* gw3 = W3 + (size_t)m * 5;
            float acc = b3[m];
#pragma unroll
            for (int o = 0; o < 5; ++o)
                acc = fmaf(h2[o], gw3[o], acc);

            out[m] = acc;
        }
    }
}

extern "C" void kernel_launch(void* const* d_in, const int* in_sizes, int n_in,
                              void* d_out, int out_size, void* d_ws, size_t ws_size,
                              hipStream_t stream) {
    const float* x  = (const float*)d_in[0];
    const float* W1 = (const float*)d_in[1];
    const float* b1 = (const float*)d_in[2];
    const float* W2 = (const float*)d_in[3];
    const float* b2 = (const float*)d_in[4];
    const float* W3 = (const float*)d_in[5];
    const float* b3 = (const float*)d_in[6];
    float* out = (float*)d_out;

    const int n = in_sizes[0];
    const int blocks = (n + MPB - 1) / MPB;

    hipLaunchKernelGGL(perturbnet_kernel, dim3(blocks), dim3(BLOCK), 0, stream,
                       x, W1, b1, W2, b2, W3, b3, out, n);
}